// architecture_28965259444741
// MI455X (gfx1250) — compile-verified
//
#include <hip/hip_runtime.h>
#include <math.h>

// ---- problem dimensions (match reference) ----
#define B_   4
#define S_   1024
#define DIN_ 768
#define D_   224
#define H_   8
#define DK_  28
#define FF_  600
#define V_   8000
#define L_   8

typedef __attribute__((ext_vector_type(2))) float v2f;
typedef __attribute__((ext_vector_type(8))) float v8f;

// D = A(16x4 f32) * B(4x16 f32) + C(16x16 f32)  -- full-precision WMMA
__device__ __forceinline__ v8f wmma_f32(v2f a, v2f b, v8f c) {
  return __builtin_amdgcn_wmma_f32_16x16x4_f32(
      /*neg_a=*/false, a, /*neg_b=*/false, b,
      /*c_mod=*/(short)0, c, /*reuse_a=*/false, /*reuse_b=*/false);
}

enum { EP_BIAS = 1, EP_RELU = 2, EP_RES = 4, EP_POSENC = 8 };

// Fully-static f32 WMMA GEMM: C[M,N] = A[M,K]*B[K,N] (+bias/resid/posenc/relu).
// One wave -> 16 x (16*NT) strip: one A frag feeds NT WMMAs on NT independent
// accumulators. All shapes/strides compile-time -> immediate-offset addressing.
template <int M, int N, int K, int LDA, int LDB, int LDC, int FLAGS, int NT>
__global__ __launch_bounds__(128) void wmma_gemm_t(
    const float* __restrict__ A, const float* __restrict__ Bm,
    const float* __restrict__ bias, const float* __restrict__ resid,
    float* __restrict__ C)
{
  const int tileM = blockIdx.y * 4 + threadIdx.y;   // wave-uniform
  const int tn0   = blockIdx.x * NT;                // first N-tile of this wave
  const int lane  = threadIdx.x;
  const int half  = lane >> 4;
  const int lr    = lane & 15;

  const int arow = tileM * 16 + lr;                 // A-frag row = lane%16
  const float* Ap = A + (size_t)arow * LDA + 2 * half;  // k = 2*half + {0,1}

  int bcol[NT];                                     // clamped for partial tiles
#pragma unroll
  for (int t = 0; t < NT; ++t) {
    const int c = (tn0 + t) * 16 + lr;
    bcol[t] = (c < N) ? c : (N - 1);
  }
  const float* Bp = Bm + (size_t)(2 * half) * LDB;

  v8f acc[NT];
#pragma unroll
  for (int t = 0; t < NT; ++t) acc[t] = {};

  for (int kb = 0; kb < K / 4; ++kb) {              // constant trip count
    v2f a; a.x = Ap[0]; a.y = Ap[1];
    Ap += 4;
    const float* bp = Bp + (size_t)kb * 4 * LDB;
#pragma unroll
    for (int t = 0; t < NT; ++t) {
      v2f b; b.x = bp[bcol[t]]; b.y = bp[LDB + bcol[t]];
      acc[t] = wmma_f32(a, b, acc[t]);              // NT independent chains
    }
  }

  // Epilogue (any divergence only after all WMMA issues)
#pragma unroll
  for (int t = 0; t < NT; ++t) {
    const int col = (tn0 + t) * 16 + lr;
    if (col >= N) continue;
    const float bv = (FLAGS & EP_BIAS) ? bias[col] : 0.0f;
#pragma unroll
    for (int r = 0; r < 8; ++r) {
      const int row = tileM * 16 + r + 8 * half;    // C-layout: M = r + 8*half
      float v = acc[t][r] + bv;
      if (FLAGS & EP_RES) v += resid[(size_t)row * LDC + col];
      if (FLAGS & EP_POSENC) {
        // ang = s / 10000^(2*d/224) = s * exp(-d * 2*ln(10000)/224)
        const int s = row % S_;
        const float ang = (float)s * __expf(-0.08223518189f * (float)col);
        v += (col & 1) ? __cosf(ang) : __sinf(ang);
      }
      if (FLAGS & EP_RELU) v = fmaxf(v, 0.0f);
      C[(size_t)row * LDC + col] = v;
    }
  }
}

// Repack Wq/Wk/Wv [L,H,D,DK] into fused [D, 3*224] (n = which*224 + h*28 + dk),
// and biases into [672].
__global__ void pack_qkv_kernel(
    const float* __restrict__ Wq, const float* __restrict__ Wk,
    const float* __restrict__ Wv, const float* __restrict__ bq,
    const float* __restrict__ bk, const float* __restrict__ bv,
    int l, float* __restrict__ Wp, float* __restrict__ bp)
{
  const int idx = blockIdx.x * blockDim.x + threadIdx.x;
  if (idx >= D_ * 3 * D_) return;
  const int d     = idx / (3 * D_);
  const int n     = idx % (3 * D_);
  const int which = n / D_;
  const int rem   = n % D_;
  const int hh    = rem / DK_;
  const int dk    = rem % DK_;
  const float* W = (which == 0) ? Wq : (which == 1) ? Wk : Wv;
  Wp[idx] = W[(((size_t)l * H_ + hh) * D_ + d) * DK_ + dk];
  if (d == 0) {
    const float* bb = (which == 0) ? bq : (which == 1) ? bk : bv;
    bp[n] = bb[((size_t)l * H_ + hh) * DK_ + dk];
  }
}

// Flash-style causal attention. qkv layout: [B,S, 3*224] (q|k|v, each h-major h*28+dk).
// One wave handles 16 query rows of one (b,h). Online softmax; P transposed via LDS.
__global__ __launch_bounds__(32) void flash_attn_kernel(
    const float* __restrict__ qkv, float* __restrict__ out)
{
  __shared__ float Pt[16][17];                 // padded to dodge bank conflicts
  const int qt   = blockIdx.x;
  const int bh   = blockIdx.y;
  const int b    = bh / H_;
  const int h    = bh % H_;
  const int lane = threadIdx.x;
  const int half = lane >> 4;
  const int lr   = lane & 15;
  const int LDQ  = 3 * D_;                     // 672

  const float* qbase = qkv + (size_t)b * S_ * LDQ + h * DK_;
  const float* kbase = qbase + D_;
  const float* vbase = qbase + 2 * D_;

  // Q fragments: M=16 query rows, K=DK=28 -> 7 k-steps of 4
  const int qrow = qt * 16 + lr;
  v2f qf[7];
#pragma unroll
  for (int kk = 0; kk < 7; ++kk) {
    const int ks = kk * 4 + 2 * half;
    qf[kk].x = qbase[(size_t)qrow * LDQ + ks];
    qf[kk].y = qbase[(size_t)qrow * LDQ + ks + 1];
  }

  v8f o0 = {}, o1 = {};
  float m[8], lsum[8];
#pragma unroll
  for (int r = 0; r < 8; ++r) { m[r] = -3.0e38f; lsum[r] = 0.0f; }
  const float scale = 0.1889822365046136f;     // 1/sqrt(28)

  for (int j = 0; j <= qt; ++j) {
    // scores tile = Q (16x28) * K^T (28x16): B[k][n] = K[s=j*16+n][k]
    v8f sc = {};
#pragma unroll
    for (int kk = 0; kk < 7; ++kk) {
      const int ks = kk * 4 + 2 * half;
      const float* kp = kbase + (size_t)(j * 16 + lr) * LDQ + ks;
      v2f bfr; bfr.x = kp[0]; bfr.y = kp[1];
      sc = wmma_f32(qf[kk], bfr, sc);
    }
    __syncthreads();                           // previous Pt reads done
#pragma unroll
    for (int r = 0; r < 8; ++r) {
      const int row = r + 8 * half;            // query row within tile
      float v = sc[r] * scale;
      if (j == qt && lr > row) v = -3.0e38f;   // causal mask on diagonal tile
      // row reductions live inside each 16-lane half-wave
      float mx = v;
      mx = fmaxf(mx, __shfl_xor(mx, 1, 32));
      mx = fmaxf(mx, __shfl_xor(mx, 2, 32));
      mx = fmaxf(mx, __shfl_xor(mx, 4, 32));
      mx = fmaxf(mx, __shfl_xor(mx, 8, 32));
      const float mn   = fmaxf(m[r], mx);
      const float corr = __expf(m[r] - mn);
      const float p    = __expf(v - mn);
      float rs = p;
      rs += __shfl_xor(rs, 1, 32);
      rs += __shfl_xor(rs, 2, 32);
      rs += __shfl_xor(rs, 4, 32);
      rs += __shfl_xor(rs, 8, 32);
      lsum[r] = lsum[r] * corr + rs;
      m[r]    = mn;
      o0[r] *= corr; o1[r] *= corr;
      Pt[row][lr] = p;                         // C-layout -> LDS (transpose stage)
    }
    __syncthreads();                           // Pt writes visible

    // O += P (16x16) * V (16x28); two N-tiles (dk 0..15, 16..27 clamped)
    const int dk2 = (16 + lr < DK_) ? (16 + lr) : (DK_ - 1);
#pragma unroll
    for (int kk = 0; kk < 4; ++kk) {
      const int ks = kk * 4 + 2 * half;
      v2f pa; pa.x = Pt[lr][ks]; pa.y = Pt[lr][ks + 1];
      const float* vp = vbase + (size_t)(j * 16 + ks) * LDQ;
      v2f b0; b0.x = vp[lr];  b0.y = vp[LDQ + lr];
      o0 = wmma_f32(pa, b0, o0);
      v2f b1; b1.x = vp[dk2]; b1.y = vp[LDQ + dk2];
      o1 = wmma_f32(pa, b1, o1);
    }
  }

  // normalize and store: out[b, q, h*28 + dk]
#pragma unroll
  for (int r = 0; r < 8; ++r) {
    const float inv = 1.0f / lsum[r];
    const int row = qt * 16 + r + 8 * half;
    float* op = out + ((size_t)(b * S_ + row)) * D_ + h * DK_;
    op[lr] = o0[r] * inv;
    if (lr < DK_ - 16) op[16 + lr] = o1[r] * inv;
  }
}

// LayerNorm over (S,D) per batch element, in-place; one block per batch elem.
__global__ __launch_bounds__(256) void ln2d_kernel(
    float* __restrict__ x, const float* __restrict__ w,
    const float* __restrict__ bsh, int l)
{
  __shared__ float red[256];
  const int ND = S_ * D_;
  const size_t base = (size_t)blockIdx.x * ND;
  const int tid = threadIdx.x;
  float s0 = 0.f, s1 = 0.f;
  for (int i = tid; i < ND; i += 256) {
    const float v = x[base + i];
    s0 += v; s1 += v * v;
  }
  red[tid] = s0; __syncthreads();
  for (int st = 128; st > 0; st >>= 1) { if (tid < st) red[tid] += red[tid + st]; __syncthreads(); }
  const float mu = red[0] / (float)ND;
  __syncthreads();
  red[tid] = s1; __syncthreads();
  for (int st = 128; st > 0; st >>= 1) { if (tid < st) red[tid] += red[tid + st]; __syncthreads(); }
  const float var  = red[0] / (float)ND - mu * mu;
  const float rstd = rsqrtf(var + 1e-5f);
  const float* wl = w   + (size_t)l * ND;
  const float* bl = bsh + (size_t)l * ND;
  for (int i = tid; i < ND; i += 256)
    x[base + i] = (x[base + i] - mu) * rstd * wl[i] + bl[i];
}

// Row softmax in place (n = V per row); one block per row.
__global__ __launch_bounds__(256) void softmax_rows_kernel(float* __restrict__ x, int n)
{
  __shared__ float red[256];
  const size_t base = (size_t)blockIdx.x * n;
  const int tid = threadIdx.x;
  float mx = -3.0e38f;
  for (int i = tid; i < n; i += 256) mx = fmaxf(mx, x[base + i]);
  red[tid] = mx; __syncthreads();
  for (int st = 128; st > 0; st >>= 1) { if (tid < st) red[tid] = fmaxf(red[tid], red[tid + st]); __syncthreads(); }
  const float m = red[0];
  __syncthreads();
  float s = 0.f;
  for (int i = tid; i < n; i += 256) s += __expf(x[base + i] - m);
  red[tid] = s; __syncthreads();
  for (int st = 128; st > 0; st >>= 1) { if (tid < st) red[tid] += red[tid + st]; __syncthreads(); }
  const float inv = 1.0f / red[0];
  for (int i = tid; i < n; i += 256) x[base + i] = __expf(x[base + i] - m) * inv;
}

extern "C" void kernel_launch(void* const* d_in, const int* in_sizes, int n_in,
                              void* d_out, int out_size, void* d_ws, size_t ws_size,
                              hipStream_t stream)
{
  (void)in_sizes; (void)n_in; (void)out_size; (void)ws_size;
  const float* x  = (const float*)d_in[0];
  const float* Wp = (const float*)d_in[1];
  const float* bp = (const float*)d_in[2];
  const float* Wq = (const float*)d_in[3];
  const float* bq = (const float*)d_in[4];
  const float* Wk = (const float*)d_in[5];
  const float* bk = (const float*)d_in[6];
  const float* Wv = (const float*)d_in[7];
  const float* bv = (const float*)d_in[8];
  const float* Wo = (const float*)d_in[9];
  const float* bo = (const float*)d_in[10];
  const float* W1 = (const float*)d_in[11];
  const float* b1 = (const float*)d_in[12];
  const float* W2 = (const float*)d_in[13];
  const float* b2 = (const float*)d_in[14];
  const float* W3 = (const float*)d_in[15];
  const float* b3 = (const float*)d_in[16];
  const float* lnw = (const float*)d_in[17];
  const float* lnb = (const float*)d_in[18];
  const float* Wf = (const float*)d_in[19];
  const float* bf = (const float*)d_in[20];
  float* out = (float*)d_out;

  // workspace carve-up (~39 MB of f32)
  float* wsf = (float*)d_ws;
  const size_t R = (size_t)B_ * S_;                       // 4096 rows
  float* h    = wsf; wsf += R * D_;                       // [4096,224]
  float* qkv  = wsf; wsf += R * 3 * D_;                   // [4096,672]
  float* aout = wsf; wsf += R * D_;                       // [4096,224]
  float* f1   = wsf; wsf += R * FF_;                      // [4096,600]
  float* f2   = wsf; wsf += R * FF_;                      // [4096,600]
  float* wpk  = wsf; wsf += (size_t)D_ * 3 * D_;          // [224,672]
  float* bpk  = wsf; wsf += 3 * D_;                       // [672]

  const dim3 blk(32, 4);
  const int MT = (int)(R / 64);                           // 64 row-strips
  constexpr int NT = 4;                                   // 4 N-tiles per wave
  auto gx = [](int N) { return ((N + 15) / 16 + NT - 1) / NT; };

  // h = x @ Wp + bp + posenc
  wmma_gemm_t<4096, D_, DIN_, DIN_, D_, D_, EP_BIAS | EP_POSENC, NT>
      <<<dim3(gx(D_), MT), blk, 0, stream>>>(x, Wp, bp, nullptr, h);

  for (int l = 0; l < L_; ++l) {
    pack_qkv_kernel<<<(D_ * 3 * D_ + 255) / 256, 256, 0, stream>>>(
        Wq, Wk, Wv, bq, bk, bv, l, wpk, bpk);
    // fused QKV projection: [4096,224] x [224,672]
    wmma_gemm_t<4096, 3 * D_, D_, D_, 3 * D_, 3 * D_, EP_BIAS, NT>
        <<<dim3(gx(3 * D_), MT), blk, 0, stream>>>(h, wpk, bpk, nullptr, qkv);
    // causal flash attention -> aout [4096,224]
    flash_attn_kernel<<<dim3(S_ / 16, B_ * H_), 32, 0, stream>>>(qkv, aout);
    // h = aout @ Wo + bo + h   (residual fused, in place)
    wmma_gemm_t<4096, D_, D_, D_, D_, D_, EP_BIAS | EP_RES, NT>
        <<<dim3(gx(D_), MT), blk, 0, stream>>>(
            aout, Wo + (size_t)l * D_ * D_, bo + (size_t)l * D_, h, h);
    ln2d_kernel<<<B_, 256, 0, stream>>>(h, lnw, lnb, l);
    // FFN
    wmma_gemm_t<4096, FF_, D_, D_, FF_, FF_, EP_BIAS | EP_RELU, NT>
        <<<dim3(gx(FF_), MT), blk, 0, stream>>>(
            h, W1 + (size_t)l * D_ * FF_, b1 + (size_t)l * FF_, nullptr, f1);
    wmma_gemm_t<4096, FF_, FF_, FF_, FF_, FF_, EP_BIAS | EP_RELU, NT>
        <<<dim3(gx(FF_), MT), blk, 0, stream>>>(
            f1, W2 + (size_t)l * FF_ * FF_, b2 + (size_t)l * FF_, nullptr, f2);
    wmma_gemm_t<4096, D_, FF_, FF_, D_, D_, EP_BIAS | EP_RES, NT>
        <<<dim3(gx(D_), MT), blk, 0, stream>>>(
            f2, W3 + (size_t)l * FF_ * D_, b3 + (size_t)l * D_, h, h);
    ln2d_kernel<<<B_, 256, 0, stream>>>(h, lnw, lnb, l);  // LN after ffn residual
    ln2d_kernel<<<B_, 256, 0, stream>>>(h, lnw, lnb, l);  // outer LN (same params)
  }

  // logits straight into d_out, softmax in place (avoids 131 MB scratch)
  wmma_gemm_t<4096, V_, D_, D_, V_, V_, EP_BIAS, NT>
      <<<dim3(gx(V_), MT), blk, 0, stream>>>(h, Wf, bf, nullptr, out);
  softmax_rows_kernel<<<(int)R, 256, 0, stream>>>(out, V_);
}